// Fusion_44315472560190
// MI455X (gfx1250) — compile-verified
//
#include <hip/hip_runtime.h>

// Problem constants (from reference)
#define BB 16
#define NN 32
#define SS 256
#define HH 768
#define H2 1536   // 2*H
#define BN (BB*NN)        // 512
#define H4 (HH/4)         // 192 float4 per row

typedef float v4f __attribute__((ext_vector_type(4)));
typedef float v2f __attribute__((ext_vector_type(2)));
typedef float v8f __attribute__((ext_vector_type(8)));

// ---------------------------------------------------------------------------
// Kernel 1: build pooled[b*N+n, 0:768] = enc[b,n,0,:],
//           pooled[b*N+n, 768:1536] = mean_{s in span} enc[b,n,s,:]
// One block per (b,n); 192 threads, each owning one float4 column group.
// ---------------------------------------------------------------------------
__global__ void k1_pool(const float* __restrict__ enc,
                        const int* __restrict__ lu,
                        float* __restrict__ pooled)
{
    const int bn = blockIdx.x;            // 0..511
    const int b  = bn / NN;
    const int t  = threadIdx.x;           // 0..191

    const int s0 = lu[2*b + 0];
    const int s1 = lu[2*b + 1];
    const float inv = 1.0f / fmaxf((float)(s1 - s0), 1.0f);

    const float* base = enc + (size_t)bn * SS * HH;
    float* pout = pooled + (size_t)bn * H2;

    // candidate pool: s = 0 row
    v4f cp = *reinterpret_cast<const v4f*>(base + t * 4);
    *reinterpret_cast<v4f*>(pout + t * 4) = cp;

    // last utterance mean over span
    v4f acc = {0.f, 0.f, 0.f, 0.f};
    for (int s = s0; s < s1; ++s) {
        v4f v = *reinterpret_cast<const v4f*>(base + (size_t)s * HH + t * 4);
        acc += v;
    }
    acc *= inv;
    *reinterpret_cast<v4f*>(pout + HH + t * 4) = acc;
}

// ---------------------------------------------------------------------------
// Kernel 2: logits = pooled @ W + bias via V_WMMA_F32_16X16X4_F32.
// GEMV M=512, K=1536, N=1.  One wave (32 lanes) per 16-row M tile.
// A layout (32-bit 16x4): lanes 0-15 M=lane: VGPR0=K0, VGPR1=K1;
//                         lanes 16-31 M=lane-16: VGPR0=K2, VGPR1=K3.
// B layout (32-bit 4x16): VGPR0: lanes0-15 row K0, lanes16-31 row K2;
//                         VGPR1: lanes0-15 row K1, lanes16-31 row K3.
//   Only column N=0 is nonzero -> lanes 0 and 16 carry W, others 0.
// D: column N=0 lives in lane 0 (M=0..7 -> c[0..7]) and lane 16 (M=8..15).
//
// W is loaded unconditionally by every lane (same address per half-wave,
// always in-bounds) and zeroed via select -> v_cndmask, no divergent branch.
// ---------------------------------------------------------------------------
__global__ void k2_logits_wmma(const float* __restrict__ pooled,
                               const float* __restrict__ W,
                               const float* __restrict__ bias,
                               float* __restrict__ logits_out)
{
    const int tile = blockIdx.x;          // 0..31 (16 rows each)
    const int lane = threadIdx.x;         // 0..31
    const int m    = lane & 15;
    const int hi   = lane >> 4;           // 0: K pair {0,1}; 1: K pair {2,3}

    const float* prow = pooled + (size_t)(tile * 16 + m) * H2;
    const float  sel  = (m == 0) ? 1.0f : 0.0f;   // lanes 0 and 16 carry W

    v8f c = {0.f, 0.f, 0.f, 0.f, 0.f, 0.f, 0.f, 0.f};

    for (int k0 = 0; k0 < H2; k0 += 4) {
        const int ka = k0 + 2 * hi;
        v2f a  = *reinterpret_cast<const v2f*>(prow + ka);  // global_load_b64
        v2f wv = *reinterpret_cast<const v2f*>(W + ka);     // global_load_b64
        v2f bm;
        bm.x = wv.x * sel;
        bm.y = wv.y * sel;
        c = __builtin_amdgcn_wmma_f32_16x16x4_f32(
                /*neg_a=*/false, a, /*neg_b=*/false, bm,
                /*c_mod=*/(short)0, c, /*reuse_a=*/false, /*reuse_b=*/false);
    }

    const float bv = bias[0];
    if (lane == 0) {
        #pragma unroll
        for (int i = 0; i < 8; ++i)
            logits_out[tile * 16 + i] = c[i] + bv;
    }
    if (lane == 16) {
        #pragma unroll
        for (int i = 0; i < 8; ++i)
            logits_out[tile * 16 + 8 + i] = c[i] + bv;
    }
}

// ---------------------------------------------------------------------------
// Kernel 3: probs = softmax over N (=32 = one wave32) per batch.
// ---------------------------------------------------------------------------
__global__ void k3_softmax(const float* __restrict__ logits,
                           float* __restrict__ probs)
{
    const int b = blockIdx.x;             // 0..15
    const int n = threadIdx.x;            // 0..31
    float x = logits[b * NN + n];

    float m = x;
    #pragma unroll
    for (int off = 16; off > 0; off >>= 1)
        m = fmaxf(m, __shfl_xor(m, off, 32));

    float e = expf(x - m);
    float ssum = e;
    #pragma unroll
    for (int off = 16; off > 0; off >>= 1)
        ssum += __shfl_xor(ssum, off, 32);

    probs[b * NN + n] = e / ssum;
}

// ---------------------------------------------------------------------------
// Kernel 4: fused[b,s,h] = sum_n probs[b,n]*masks[b,n,s]*enc[b,n,s,h]
// One block per (b,s); 192 threads each own one float4 of H.
// Streams all 403 MB of enc exactly once -> HBM bound.  No reuse of either
// the streamed input or the output (12.6 MB, write-once), so use
// non-temporal loads/stores to keep L2 clean.
// ---------------------------------------------------------------------------
__global__ void k4_fuse(const float* __restrict__ enc,
                        const float* __restrict__ masks,
                        const float* __restrict__ probs,
                        float* __restrict__ fused)
{
    const int bs = blockIdx.x;            // 0..4095
    const int b  = bs / SS;
    const int s  = bs % SS;
    const int t  = threadIdx.x;           // 0..191

    __shared__ float wsm[NN];
    if (t < NN)
        wsm[t] = probs[b * NN + t] * masks[(size_t)(b * NN + t) * SS + s];
    __syncthreads();

    const float* encb = enc + ((size_t)b * NN) * SS * HH + (size_t)s * HH;

    v4f acc = {0.f, 0.f, 0.f, 0.f};
    #pragma unroll 4
    for (int n = 0; n < NN; ++n) {
        const float w = wsm[n];
        v4f v = __builtin_nontemporal_load(
            reinterpret_cast<const v4f*>(encb + (size_t)n * SS * HH + t * 4));
        acc += w * v;
    }

    __builtin_nontemporal_store(
        acc, reinterpret_cast<v4f*>(fused + (size_t)bs * HH + t * 4));
}

// ---------------------------------------------------------------------------
extern "C" void kernel_launch(void* const* d_in, const int* in_sizes, int n_in,
                              void* d_out, int out_size, void* d_ws, size_t ws_size,
                              hipStream_t stream)
{
    const float* enc   = (const float*)d_in[0];   // (B,N,S,H)
    const float* masks = (const float*)d_in[1];   // (B,N,S)
    const int*   lu    = (const int*)  d_in[2];   // (B,2)
    const float* W     = (const float*)d_in[3];   // (2H,1)
    const float* bias  = (const float*)d_in[4];   // (1,)

    float* out    = (float*)d_out;
    float* probs  = out;                 // 512
    float* logits = out + BN;            // 512
    float* fused  = out + 2 * BN;        // B*S*H

    float* pooled = (float*)d_ws;        // 512 * 1536 floats = 3 MB

    k1_pool        <<<dim3(BN),      dim3(H4), 0, stream>>>(enc, lu, pooled);
    k2_logits_wmma <<<dim3(BN / 16), dim3(32), 0, stream>>>(pooled, W, bias, logits);
    k3_softmax     <<<dim3(BB),      dim3(NN), 0, stream>>>(logits, probs);
    k4_fuse        <<<dim3(BB * SS), dim3(H4), 0, stream>>>(enc, masks, probs, fused);
}